// Until_14577119003478
// MI455X (gfx1250) — compile-verified
//
#include <hip/hip_runtime.h>
#include <hip/hip_bf16.h>
#include <stdint.h>

// Problem constants (reference: B=1024, T=8192, C=2, float32)
#define T_LEN   8192
#define TS      1024                       // t-steps per LDS stage
#define NST     (T_LEN / TS)               // 8 stages
#define DW      (TS * 2)                   // dwords per array per stage (2048)
#define PDW     (DW + ((DW >> 6) << 2))    // padded: +4 dwords per 64 (2176)
#define NCHUNK  128                        // chunks per channel per stage
#define CHUNK   (TS / NCHUNK)              // 8 t-steps per thread-chunk
#define NTHR    256                        // 8 waves (wave32)

__device__ __forceinline__ int padi(int d) { return d + ((d >> 6) << 2); }

// ---- CDNA5 async global<->LDS path (guarded; fallback is synchronous) ----
#if __has_builtin(__builtin_amdgcn_global_load_async_to_lds_b128) && \
    __has_builtin(__builtin_amdgcn_global_store_async_from_lds_b128)
#define HAVE_ASYNC 1
typedef int v4i __attribute__((vector_size(16)));
typedef __attribute__((address_space(1))) v4i* gv4i_p;   // global pointee
typedef __attribute__((address_space(3))) v4i* lv4i_p;   // LDS pointee
__device__ __forceinline__ void async_ld16(const float* g, float* l) {
  __builtin_amdgcn_global_load_async_to_lds_b128((gv4i_p)(void*)g,
                                                 (lv4i_p)(void*)l, 0, 0);
}
__device__ __forceinline__ void async_st16(float* g, const float* l) {
  __builtin_amdgcn_global_store_async_from_lds_b128((gv4i_p)(void*)g,
                                                    (lv4i_p)(void*)l, 0, 0);
}
#else
#define HAVE_ASYNC 0
__device__ __forceinline__ void async_ld16(const float* g, float* l) {
  *(float4*)l = *(const float4*)g;
}
__device__ __forceinline__ void async_st16(float* g, const float* l) {
  *(float4*)g = *(const float4*)l;
}
#endif

__device__ __forceinline__ void wait_async() {
#if __has_builtin(__builtin_amdgcn_s_wait_asynccnt)
  __builtin_amdgcn_s_wait_asynccnt(0);
#elif HAVE_ASYNC
  asm volatile("s_wait_asynccnt 0" ::: "memory");
#endif
}

// r[t] = max(min(1, psi[t]), min(phi[t], r[t+1])), r[T] = -inf
// Step functions f_t(x) = max(v, min(p, x)) are closed under composition:
//   (A,B) o (A2,B2) = (max(A, min(B, A2)), min(B, B2))   [exact, same semiring]
__global__ __launch_bounds__(NTHR) void until_kernel(
    const float* __restrict__ phi, const float* __restrict__ psi,
    float* __restrict__ out) {
  __shared__ float sPhi[2][PDW];
  __shared__ float sPsi[2][PDW];
  __shared__ float sA[2][NCHUNK];
  __shared__ float sB[2][NCHUNK];
  __shared__ float sCr[2];

  const int tid = threadIdx.x;
  const int c   = tid >> 7;            // channel 0/1
  const int k   = tid & (NCHUNK - 1);  // chunk index within channel
  const long long base = (long long)blockIdx.x * (T_LEN * 2);
  const float NEG = -__builtin_inff();
  const float POS =  __builtin_inff();

  if (tid < 2) sCr[tid] = NEG;         // r[T] = -inf

  // Prologue: async-load the last stage into buffer 0 (B128, padded swizzle)
  {
    const float* gp = phi + base + (long long)(NST - 1) * DW;
    const float* gq = psi + base + (long long)(NST - 1) * DW;
    for (int i = 0; i < DW / (NTHR * 4); ++i) {
      const int d = (tid + i * NTHR) * 4;
      async_ld16(gp + d, &sPhi[0][padi(d)]);
      async_ld16(gq + d, &sPsi[0][padi(d)]);
    }
  }

  int p = 0;
  for (int s = NST - 1; s >= 0; --s) {
    wait_async();        // my loads (and my prior stage's stores) complete
    __syncthreads();     // => all waves' LDS writes visible; buffers safe

    // Prefetch the next (earlier-t) stage into the other buffer
    if (s > 0) {
      const float* gp = phi + base + (long long)(s - 1) * DW;
      const float* gq = psi + base + (long long)(s - 1) * DW;
      for (int i = 0; i < DW / (NTHR * 4); ++i) {
        const int d = (tid + i * NTHR) * 4;
        async_ld16(gp + d, &sPhi[p ^ 1][padi(d)]);
        async_ld16(gq + d, &sPsi[p ^ 1][padi(d)]);
      }
    }

    // (1) Compose this thread's chunk function, descending t
    float A = NEG, Bv = POS;             // identity: f(x)=x
    const int t0 = k * CHUNK;
    for (int j = CHUNK - 1; j >= 0; --j) {
      const int pd  = padi((t0 + j) * 2 + c);
      const float pv = sPhi[p][pd];
      const float qv = sPsi[p][pd];
      A  = fmaxf(fminf(1.0f, qv), fminf(pv, A));
      Bv = fminf(pv, Bv);
    }
    sA[c][k] = A; sB[c][k] = Bv;
    __syncthreads();

    // (2) Reverse inclusive scan of chunk functions (Hillis-Steele, 7 steps)
    for (int dstep = 1; dstep < NCHUNK; dstep <<= 1) {
      float A2 = NEG, B2 = POS;
      if (k + dstep < NCHUNK) { A2 = sA[c][k + dstep]; B2 = sB[c][k + dstep]; }
      __syncthreads();
      A  = fmaxf(A, fminf(Bv, A2));
      Bv = fminf(Bv, B2);
      sA[c][k] = A; sB[c][k] = Bv;
      __syncthreads();
    }

    // (3) Carry entering this chunk's last element: G_{k+1}(stage_carry)
    const float cr = sCr[c];
    float r = (k == NCHUNK - 1) ? cr
            : fmaxf(sA[c][k + 1], fminf(sB[c][k + 1], cr));
    __syncthreads();     // everyone consumed sCr before k==0 overwrites it

    // (4) Second pass: emit outputs in place over the psi buffer
    for (int j = CHUNK - 1; j >= 0; --j) {
      const int pd  = padi((t0 + j) * 2 + c);
      const float pv = sPhi[p][pd];
      const float qv = sPsi[p][pd];
      r = fmaxf(fminf(1.0f, qv), fminf(pv, r));
      sPsi[p][pd] = r;
    }
    if (k == 0) sCr[c] = r;   // new stage carry = r at stage start
    __syncthreads();          // psi buffer complete; sCr visible next stage

    // (5) Async-store outputs LDS -> global (coalesced B128)
    {
      float* go = out + base + (long long)s * DW;
      for (int i = 0; i < DW / (NTHR * 4); ++i) {
        const int d = (tid + i * NTHR) * 4;
        async_st16(go + d, &sPsi[p][padi(d)]);
      }
    }
    p ^= 1;
  }
  // s_endpgm performs an implicit wait-idle, draining the final async stores.
}

extern "C" void kernel_launch(void* const* d_in, const int* in_sizes, int n_in,
                              void* d_out, int out_size, void* d_ws, size_t ws_size,
                              hipStream_t stream) {
  const float* phi = (const float*)d_in[0];
  const float* psi = (const float*)d_in[1];
  float* out = (float*)d_out;
  const int Bsz = in_sizes[0] / (T_LEN * 2);   // 1024
  until_kernel<<<dim3(Bsz), dim3(NTHR), 0, stream>>>(phi, psi, out);
}